// UniCORNNLayer_27075473834253
// MI455X (gfx1250) — compile-verified
//
#include <hip/hip_runtime.h>
#include <hip/hip_bf16.h>
#include <math.h>

typedef __attribute__((ext_vector_type(2))) float v2f;
typedef __attribute__((ext_vector_type(4))) float v4f;
typedef __attribute__((ext_vector_type(8))) float v8f;
typedef __attribute__((ext_vector_type(4))) int   v4i;

#define UNITS   512
#define FT_DIM  128
#define OUT_DIM 10
#define BATCH   64
#define SEQ     512
#define EPS     0.03f
#define ALPHA_C 0.9f

// states layout in d_out: [t][b][ y1n(0:512) | z1n(512:1024) | y2n(1024:1536) | z2n(1536:2048) ]
#define ROWST   2048
#define STEPST  (BATCH * ROWST)
#define STATES_TOTAL ((size_t)SEQ * STEPST)

#define NWG      16                   // persistent workgroups (one per WGP, all resident)
#define A_STRIDE 516                  // 512 + 4 pad: conflict-free b64 LDS reads, 16B-aligned rows
#define A_LDS_FLOATS (BATCH * A_STRIDE)       // 33024 floats = 129 KB
#define B_LDS_FLOATS (2 * 128 * 32 * 2)       // 16384 floats = 64 KB (pre-swizzled B fragments)
#define LDS_BYTES ((A_LDS_FLOATS + B_LDS_FLOATS) * 4)  // 197632 B < 320 KB/WGP

// CDNA5 async global->LDS path (guarded; falls back to b128 copy via VGPRs)
#if defined(__has_builtin)
#if __has_builtin(__builtin_amdgcn_global_load_async_to_lds_b128) && \
    __has_builtin(__builtin_amdgcn_s_wait_asynccnt)
#define HAVE_ASYNC_LDS 1
#endif
#endif

#if defined(HAVE_ASYNC_LDS)
typedef __attribute__((address_space(1))) v4i gv4i;   // global int4
typedef __attribute__((address_space(3))) v4i lv4i;   // LDS int4
#endif

__device__ __forceinline__ float mod_tanh_scale(float c) {
    return EPS * (0.5f + 0.5f * tanhf(0.5f * c));   // EPSILON * (0.5 + 0.5*tanh(c/2))
}

// ------------------------------------------------------------------
// XV1[t][b][:] = inputs[b][t][:] @ V1, stored into y1n slot of states.
// One wave32 per 16x16 tile, V_WMMA_F32_16X16X4_F32, K = 128.
// ------------------------------------------------------------------
__global__ void unicornn_xv1_kernel(const float* __restrict__ inp,
                                    const float* __restrict__ V1,
                                    float* __restrict__ S) {
    const int gwave = (blockIdx.x * blockDim.x + threadIdx.x) >> 5;
    const int lane  = threadIdx.x & 31;
    const int half  = lane >> 4;
    const int l     = lane & 15;

    const int mt = gwave >> 5;        // 0..2047
    const int nt = gwave & 31;        // 0..31
    const int m0 = mt * 16;
    const int n0 = nt * 16;

    const int am = m0 + l;            // m = t*64 + b
    const int at = am >> 6;
    const int ab = am & 63;
    const float* __restrict__ arow = inp + ((size_t)ab * SEQ + at) * FT_DIM;

    v8f acc = {};
    for (int k0 = 0; k0 < FT_DIM; k0 += 4) {
        v2f a;
        a.x = arow[k0 + 2 * half];
        a.y = arow[k0 + 2 * half + 1];
        v2f b;
        b.x = V1[(size_t)(k0 + 2 * half) * UNITS + n0 + l];
        b.y = V1[(size_t)(k0 + 2 * half + 1) * UNITS + n0 + l];
        acc = __builtin_amdgcn_wmma_f32_16x16x4_f32(
            false, a, false, b, (short)0, acc, false, false);
    }
#pragma unroll
    for (int r = 0; r < 8; ++r) {
        const int m = m0 + r + 8 * half;
        S[(size_t)m * ROWST + n0 + l] = acc[r];
    }
}

// ------------------------------------------------------------------
// Persistent scan kernel: all 512 recurrent steps in ONE launch.
// NWG=16 workgroups x 256 threads (8 waves). Workgroup g owns unit
// columns [32g, 32g+32): y1/z1/y2/z2 carried in registers; only y1n
// is exchanged through global memory + device-wide spin barrier.
// GEMM h2pre = y1n[t] @ V2[:, own 32 cols] runs out of LDS, with the
// A tile staged via CDNA5 async global->LDS loads (ASYNCcnt).
// ------------------------------------------------------------------
__global__ void unicornn_scan_kernel(const float* __restrict__ w1,
                                     const float* __restrict__ w2,
                                     const float* __restrict__ V2,
                                     const float* __restrict__ b1,
                                     const float* __restrict__ b2,
                                     const float* __restrict__ c1,
                                     const float* __restrict__ c2,
                                     float* __restrict__ S,
                                     unsigned* __restrict__ bar_ctr) {
    extern __shared__ float lds[];
    float* Alds = lds;                     // [64][516]  y1n[t] staged per step
    float* Blds = lds + A_LDS_FLOATS;      // [2][128][32][2] pre-swizzled V2 fragments

    const int tid  = threadIdx.x;
    const int g    = blockIdx.x;           // 0..15
    const int wv   = tid >> 5;             // 0..7
    const int lane = tid & 31;
    const int half = lane >> 4;
    const int l    = lane & 15;
    const int mt   = wv >> 1;              // 0..3 batch tile
    const int ntl  = wv & 1;               // 0..1 local unit tile
    const int u    = g * 32 + ntl * 16 + l;   // layer-2 column owned by this lane

    // ---- stage V2 block once, swizzled into per-lane B-fragment order ----
    for (int e = tid; e < 2 * 128 * 32; e += 256) {
        const int entl = e >> 12;          // /4096
        const int ekb  = (e >> 5) & 127;
        const int eln  = e & 31;
        const int k    = 4 * ekb + 2 * (eln >> 4);
        const int ncol = g * 32 + entl * 16 + (eln & 15);
        v2f bv;
        bv.x = V2[(size_t)k * UNITS + ncol];
        bv.y = V2[(size_t)(k + 1) * UNITS + ncol];
        *(v2f*)(Blds + 2 * e) = bv;
    }

    // ---- per-thread layer-1 state/constants: idx = tid + 256*i -> (b, uu) ----
    float y1[8], z1[8], d1c[8], w1c[8], b1c[8];
#pragma unroll
    for (int i = 0; i < 8; ++i) {
        const int idx  = tid + 256 * i;
        const int uu   = idx & 31;
        const int ucol = g * 32 + uu;
        d1c[i] = mod_tanh_scale(c1[ucol]);
        w1c[i] = w1[ucol];
        b1c[i] = b1[ucol];
        y1[i] = 0.0f;
        z1[i] = 0.0f;
    }
    // ---- per-lane layer-2 state/constants (column u, 8 batch rows) ----
    const float d2  = mod_tanh_scale(c2[u]);
    const float w2u = w2[u];
    const float b2u = b2[u];
    float y2[8], z2[8];
#pragma unroll
    for (int r = 0; r < 8; ++r) { y2[r] = 0.0f; z2[r] = 0.0f; }

    for (int t = 0; t < SEQ; ++t) {
        float* St = S + (size_t)t * STEPST;

        // (1) layer-1 leapfrog for own columns; publish y1n[t], z1n[t];
        //     prefetch next step's XV1 slice toward this WGP.
#pragma unroll
        for (int i = 0; i < 8; ++i) {
            const int idx = tid + 256 * i;
            const int b   = idx >> 5;
            const int uu  = idx & 31;
            float* row = St + (size_t)b * ROWST;
            const int ucol = g * 32 + uu;
            const float xv = row[ucol];                     // XV1[t][b][ucol]
            const float h1 = tanhf(xv + w1c[i] * y1[i] + b1c[i]);
            z1[i] = z1[i] - d1c[i] * (h1 + ALPHA_C * y1[i]);
            y1[i] = y1[i] + d1c[i] * z1[i];
            row[ucol]       = y1[i];
            row[512 + ucol] = z1[i];
            if (t + 1 < SEQ) {
                __builtin_prefetch(row + STEPST + ucol, 0, 1);  // global_prefetch_b8
            }
        }

        // (2) device-wide barrier: release y1n[t], acquire others' writes
        __threadfence();
        __syncthreads();
        if (tid == 0) {
            __hip_atomic_fetch_add(bar_ctr, 1u, __ATOMIC_RELEASE,
                                   __HIP_MEMORY_SCOPE_AGENT);
            const unsigned target = (unsigned)(NWG * (t + 1));
            while (__hip_atomic_load(bar_ctr, __ATOMIC_ACQUIRE,
                                     __HIP_MEMORY_SCOPE_AGENT) < target) {
                __builtin_amdgcn_s_sleep(2);
            }
        }
        __syncthreads();
        __threadfence();

        // (3) stage full y1n[t] (64x512) into LDS: async global->LDS b128
        for (int ch = tid; ch < BATCH * 128; ch += 256) {
            const int row = ch >> 7;
            const int c4  = (ch & 127) << 2;
            const float* gsrc = St + (size_t)row * ROWST + c4;
            float* ldst = Alds + row * A_STRIDE + c4;
#if defined(HAVE_ASYNC_LDS)
            __builtin_amdgcn_global_load_async_to_lds_b128(
                (gv4i*)(v4i*)(float*)gsrc,
                (lv4i*)(v4i*)ldst,
                0, 0);
#else
            *(v4f*)ldst = *(const v4f*)gsrc;
#endif
        }
#if defined(HAVE_ASYNC_LDS)
        __builtin_amdgcn_s_wait_asynccnt(0);
#endif
        __syncthreads();

        // (4) WMMA GEMM for tile (mt, ntl), entirely from LDS
        v8f acc = {};
        const float* __restrict__ Abase =
            Alds + (mt * 16 + l) * A_STRIDE + 2 * half;
        const float* __restrict__ Bbase =
            Blds + ((size_t)ntl * 128 * 32 + lane) * 2;
#pragma unroll 8
        for (int kb = 0; kb < 128; ++kb) {
            v2f a = *(const v2f*)(Abase + 4 * kb);
            v2f b = *(const v2f*)(Bbase + 64 * kb);
            acc = __builtin_amdgcn_wmma_f32_16x16x4_f32(
                false, a, false, b, (short)0, acc, false, false);
        }

        // (5) layer-2 leapfrog in registers; publish y2n[t], z2n[t]
#pragma unroll
        for (int r = 0; r < 8; ++r) {
            const int b = mt * 16 + r + 8 * half;
            const float h2 = tanhf(acc[r] + w2u * y2[r] + b2u);
            z2[r] = z2[r] - d2 * (h2 + ALPHA_C * y2[r]);
            y2[r] = y2[r] + d2 * z2[r];
            float* row = St + (size_t)b * ROWST;
            row[1024 + u] = y2[r];
            row[1536 + u] = z2[r];
        }
        // next iteration's barrier (2) separates these LDS reads from restaging
    }
}

// ------------------------------------------------------------------
// Head: Y = softmax(states[SEQ-1] @ D + bias_out). 64 threads.
// ------------------------------------------------------------------
__global__ void unicornn_head_kernel(const float* __restrict__ D,
                                     const float* __restrict__ bias_out,
                                     const float* __restrict__ S,
                                     float* __restrict__ Y) {
    const int b = threadIdx.x;
    if (b >= BATCH) return;
    float acc[OUT_DIM];
#pragma unroll
    for (int j = 0; j < OUT_DIM; ++j) acc[j] = bias_out[j];
    const float* __restrict__ srow =
        S + (size_t)(SEQ - 1) * STEPST + (size_t)b * ROWST;
    for (int k = 0; k < 4 * UNITS; ++k) {
        const float sv = srow[k];
        const float* __restrict__ drow = D + (size_t)k * OUT_DIM;
#pragma unroll
        for (int j = 0; j < OUT_DIM; ++j) acc[j] += sv * drow[j];
    }
    float m = acc[0];
#pragma unroll
    for (int j = 1; j < OUT_DIM; ++j) m = fmaxf(m, acc[j]);
    float sum = 0.0f;
#pragma unroll
    for (int j = 0; j < OUT_DIM; ++j) {
        acc[j] = __expf(acc[j] - m);
        sum += acc[j];
    }
    const float inv = 1.0f / sum;
#pragma unroll
    for (int j = 0; j < OUT_DIM; ++j) Y[(size_t)b * OUT_DIM + j] = acc[j] * inv;
}

extern "C" void kernel_launch(void* const* d_in, const int* in_sizes, int n_in,
                              void* d_out, int out_size, void* d_ws, size_t ws_size,
                              hipStream_t stream) {
    const float* inputs   = (const float*)d_in[0];
    const float* w1       = (const float*)d_in[1];
    const float* w2       = (const float*)d_in[2];
    const float* V1       = (const float*)d_in[3];
    const float* V2       = (const float*)d_in[4];
    const float* b1       = (const float*)d_in[5];
    const float* b2       = (const float*)d_in[6];
    const float* c1       = (const float*)d_in[7];
    const float* c2       = (const float*)d_in[8];
    const float* Dw       = (const float*)d_in[9];
    const float* bias_out = (const float*)d_in[10];

    float* S = (float*)d_out;                 // states [SEQ][BATCH][2048]
    float* Y = S + STATES_TOTAL;              // softmax output [BATCH][10]

    // zero the spin-barrier counter (graph-capturable, deterministic per call)
    (void)hipMemsetAsync(d_ws, 0, 64, stream);

    // 1) XV1 for all timesteps (parallel WMMA GEMM)
    unicornn_xv1_kernel<<<8192, 256, 0, stream>>>(inputs, V1, S);

    // 2) entire 512-step recurrence in one persistent kernel
    unicornn_scan_kernel<<<NWG, 256, LDS_BYTES, stream>>>(
        w1, w2, V2, b1, b2, c1, c2, S, (unsigned*)d_ws);

    // 3) classification head + softmax
    unicornn_head_kernel<<<1, 64, 0, stream>>>(Dw, bias_out, S, Y);
}